// GCN_dgl_36129264894559
// MI455X (gfx1250) — compile-verified
//
#include <hip/hip_runtime.h>

#define NFEAT 96
#define WPITCH 98   // transposed-W LDS pitch: even (8B-aligned b64 loads) and
                    // bank-step 34 mod 64 -> 16 lanes hit 16 distinct even-bank
                    // pairs, upper half-wave hits the odd pairs: conflict-free.

typedef __attribute__((ext_vector_type(2))) float v2f;
typedef __attribute__((ext_vector_type(8))) float v8f;

// ---------------- utility: zero a float region ----------------
__global__ void k_zero(float* __restrict__ p, int n) {
  int i = blockIdx.x * blockDim.x + threadIdx.x;
  if (i < n) p[i] = 0.0f;
}

// ---------------- degree histograms (float atomics, L2-resident) ----------------
__global__ void k_degrees(const int* __restrict__ src, const int* __restrict__ dst,
                          float* __restrict__ deg_out, float* __restrict__ deg_in, int E) {
  int e = blockIdx.x * blockDim.x + threadIdx.x;
  if (e < E) {
    atomicAdd(&deg_out[src[e]], 1.0f);
    atomicAdd(&deg_in[dst[e]], 1.0f);
  }
}

// ---------------- deg -> rsqrt(max(deg,1)) in place ----------------
__global__ void k_norm(float* __restrict__ deg, int N) {
  int i = blockIdx.x * blockDim.x + threadIdx.x;
  if (i < N) {
    float d = deg[i];
    deg[i] = rsqrtf(d > 0.0f ? d : 1.0f);
  }
}

// ---------------- SpMM scatter: agg[dst] += h[src] * norm_src[src] ----------------
// One thread per (edge, feature). Feature matrix (19.2 MB) is L2-resident on the
// 192 MB L2; scatter-adds resolve as global_atomic_add_f32 at L2.
__global__ void k_spmm(const float* __restrict__ h, const int* __restrict__ src,
                       const int* __restrict__ dst, const float* __restrict__ ns,
                       float* __restrict__ agg, int E) {
  unsigned i = blockIdx.x * blockDim.x + threadIdx.x;
  unsigned total = (unsigned)E * NFEAT;
  if (i >= total) return;
  int e = (int)(i / NFEAT);
  int f = (int)(i % NFEAT);
  int s = src[e];
  int d = dst[e];
  atomicAdd(&agg[(long long)d * NFEAT + f],
            h[(long long)s * NFEAT + f] * ns[s]);
}

// ---------------- fused (A .* norm_dst) @ W + b, ReLU  via V_WMMA_F32_16X16X4_F32 ----
// Block = 256 threads = 8 wave32s. Each wave computes a 16-row x 96-col output tile.
// W is staged TRANSPOSED in LDS (ldsWT[c*WPITCH + k]) so each B fragment is one
// aligned, conflict-free ds_load_b64. All 24 A fragments are preloaded and scaled
// before the K loop, so the WMMA loop issues only DS loads + WMMAs.
// Row indices are clamped (never predicated) so EXEC stays all-ones around WMMA.
__global__ __launch_bounds__(256) void k_gemm_relu(
    const float* __restrict__ A, const float* __restrict__ ndst,
    const float* __restrict__ W, const float* __restrict__ bias,
    float* __restrict__ out, int N) {
  __shared__ float ldsWT[NFEAT * WPITCH];

  for (int idx = threadIdx.x; idx < NFEAT * NFEAT; idx += 256) {
    int k = idx / NFEAT, c = idx % NFEAT;
    ldsWT[c * WPITCH + k] = W[idx];           // transpose on the way in
  }
  __syncthreads();

  const int lane = threadIdx.x & 31;
  const int wv   = threadIdx.x >> 5;
  const int row0 = blockIdx.x * 128 + wv * 16;

  // A-fragment layout (16x4 f32): lanes 0-15 -> M=lane,    K={k,k+1};
  //                               lanes 16-31 -> M=lane-16, K={k+2,k+3}.
  const int m  = lane & 15;
  const int kh = (lane >> 4) << 1;   // 0 or 2

  int arow = row0 + m;
  if (arow > N - 1) arow = N - 1;    // clamp, keep EXEC uniform
  const float scale = ndst[arow];
  const float* __restrict__ Ar = A + (long long)arow * NFEAT + kh;

  // Preload all 24 A fragments (24 x b64 loads, one wait), then one scale pass.
  v2f af[24];
#pragma unroll
  for (int t = 0; t < 24; ++t)
    af[t] = *(const v2f*)(Ar + 4 * t);
#pragma unroll
  for (int t = 0; t < 24; ++t)
    af[t] *= scale;                  // v_pk_mul_f32

  v8f acc[6];
  const v8f vzero = {};
#pragma unroll
  for (int n = 0; n < 6; ++n) acc[n] = vzero;

  // Per lane: B fragment column c = n*16 + m, K pair = k + kh.
  const float* __restrict__ Bbase = &ldsWT[m * WPITCH + kh];

#pragma unroll
  for (int t = 0; t < 24; ++t) {
    const int k = 4 * t;
#pragma unroll
    for (int n = 0; n < 6; ++n) {
      v2f b = *(const v2f*)(Bbase + n * (16 * WPITCH) + k);  // one ds_load_b64
      acc[n] = __builtin_amdgcn_wmma_f32_16x16x4_f32(
          false, af[t], false, b, (short)0, acc[n], false, false);
    }
  }

  // C/D layout: VGPR r -> row (row0 + r + 8*(lane>=16)), col = lane&15 within N-tile.
  const int rofs = (lane >> 4) << 3;
#pragma unroll
  for (int n = 0; n < 6; ++n) {
    const int c = n * 16 + m;
    const float bb = bias[c];
#pragma unroll
    for (int r = 0; r < 8; ++r) {
      int row = row0 + r + rofs;
      if (row < N) {
        float v = acc[n][r] + bb;
        out[(long long)row * NFEAT + c] = v > 0.0f ? v : 0.0f;
      }
    }
  }
}

// ---------------- column max over rows (ReLU output => >=0, int-bit atomicMax OK) ----
__global__ void k_colmax(const float* __restrict__ h, int* __restrict__ gmax, int N) {
  const int f = threadIdx.x;           // blockDim.x == 96
  float mx = 0.0f;
  for (int row = blockIdx.x; row < N; row += gridDim.x)
    mx = fmaxf(mx, h[(long long)row * NFEAT + f]);
  atomicMax(&gmax[f], __float_as_int(mx));
}

// ---------------- final: out[c] = bl[c] + sum_k g[k] * Wl[k,c] ----------------
__global__ void k_final(const int* __restrict__ gmax, const float* __restrict__ Wl,
                        const float* __restrict__ bl, float* __restrict__ out) {
  int c = threadIdx.x;
  if (c < 2) {
    float acc = bl[c];
    for (int k = 0; k < NFEAT; ++k)
      acc += __int_as_float(gmax[k]) * Wl[k * 2 + c];
    out[c] = acc;
  }
}

extern "C" void kernel_launch(void* const* d_in, const int* in_sizes, int n_in,
                              void* d_out, int out_size, void* d_ws, size_t ws_size,
                              hipStream_t stream) {
  const float* x   = (const float*)d_in[0];
  const int*   src = (const int*)d_in[1];
  const int*   dst = (const int*)d_in[2];
  const float* W1  = (const float*)d_in[3];
  const float* b1  = (const float*)d_in[4];
  const float* W2  = (const float*)d_in[5];
  const float* b2  = (const float*)d_in[6];
  const float* Wl  = (const float*)d_in[7];
  const float* bl  = (const float*)d_in[8];
  float* out = (float*)d_out;

  const int N = in_sizes[0] / NFEAT;   // 50000
  const int E = in_sizes[1];           // 800000

  // workspace layout (floats): [bufA N*96][norm_src N][norm_dst N][gmax 128][bufB N*96]
  float* bufA = (float*)d_ws;
  float* nsrc = bufA + (size_t)N * NFEAT;
  float* ndst = nsrc + N;
  int*   gmax = (int*)(ndst + N);
  float* bufB = (float*)(gmax + 128);

  const int zcount = N * NFEAT + 2 * N + 128;   // bufA + degs + gmax (contiguous)
  const unsigned total = (unsigned)E * NFEAT;
  const int gblocks = (N + 127) / 128;

  k_zero   <<<(zcount + 255) / 256, 256, 0, stream>>>(bufA, zcount);
  k_degrees<<<(E + 255) / 256,      256, 0, stream>>>(src, dst, nsrc, ndst, E);
  k_norm   <<<(N + 255) / 256,      256, 0, stream>>>(nsrc, N);
  k_norm   <<<(N + 255) / 256,      256, 0, stream>>>(ndst, N);

  // layer 1
  k_spmm     <<<(total + 255) / 256, 256, 0, stream>>>(x, src, dst, nsrc, bufA, E);
  k_gemm_relu<<<gblocks, 256, 0, stream>>>(bufA, ndst, W1, b1, bufB, N);

  // layer 2
  k_zero     <<<(N * NFEAT + 255) / 256, 256, 0, stream>>>(bufA, N * NFEAT);
  k_spmm     <<<(total + 255) / 256, 256, 0, stream>>>(bufB, src, dst, nsrc, bufA, E);
  k_gemm_relu<<<gblocks, 256, 0, stream>>>(bufA, ndst, W2, b2, bufB, N);

  // readout + classifier
  k_colmax<<<256, NFEAT, 0, stream>>>(bufB, gmax, N);
  k_final <<<1, 32, 0, stream>>>(gmax, Wl, bl, out);
}